// Router_4123168604833
// MI455X (gfx1250) — compile-verified
//
#include <hip/hip_runtime.h>
#include <hip/hip_bf16.h>

// MoE router for MI455X (gfx1250, wave32).
//   out = [expert_indices i32 E*cap | expert_weights f32 E*cap | w_flat f32 T*2 | logits f32 T*64]
// Kernel 1: f32 WMMA GEMM (V_WMMA_F32_16X16X4_F32). A-activations are staged into
//           double-buffered LDS with GLOBAL_LOAD_ASYNC_TO_LDS_B128 (ASYNCcnt-tracked)
//           when the toolchain exposes the builtin; otherwise direct global b64 loads.
// Kernel 2: deterministic slot-major greedy capacity assignment (wave32 ballot scan).

typedef __attribute__((ext_vector_type(2))) float v2f;
typedef __attribute__((ext_vector_type(4))) float v4f;
typedef __attribute__((ext_vector_type(8))) float v8f;
typedef __attribute__((ext_vector_type(4))) int   v4i;

#define GLOBAL_AS __attribute__((address_space(1)))
#define LDS_AS    __attribute__((address_space(3)))

#define HDIM 4096
#define NEXP 64
#define TOK_PER_WAVE 16
#define WAVES_PER_BLK 8
#define TOK_PER_BLK (TOK_PER_WAVE * WAVES_PER_BLK)   // 128
#define LDS_STRIDE 65                                 // 64 + 1 pad: conflict-free rows
#define KCHUNK 64                                     // K-floats staged per async chunk

#if defined(__has_builtin)
#  if __has_builtin(__builtin_amdgcn_global_load_async_to_lds_b128)
#    define HAVE_ASYNC_LDS 1
#  else
#    define HAVE_ASYNC_LDS 0
#  endif
#  if __has_builtin(__builtin_amdgcn_s_wait_asynccnt)
#    define WAIT_ASYNC(n) __builtin_amdgcn_s_wait_asynccnt((unsigned short)(n))
#  else
#    define WAIT_ASYNC(n) asm volatile("s_wait_asynccnt %0" ::"i"(n))
#  endif
#else
#  define HAVE_ASYNC_LDS 0
#  define WAIT_ASYNC(n) asm volatile("s_wait_asynccnt %0" ::"i"(n))
#endif

__global__ __launch_bounds__(256) void router_gemm_topk_kernel(
    const float* __restrict__ hidden,    // [T, HDIM]
    const float* __restrict__ Wr,        // [NEXP, HDIM]
    float* __restrict__ out_logits,      // [T, NEXP]
    float* __restrict__ out_wflat,       // [T, 2]
    int* __restrict__ idx_ws)            // [T, 2]
{
    // Buffer 0 doubles as the logits staging buffer for the epilogue.
    __shared__ float smem[2][TOK_PER_BLK * LDS_STRIDE];

    const int tid     = threadIdx.x;
    const int wave    = tid >> 5;
    const int lane    = tid & 31;
    const int lane_lo = lane & 15;
    const int hi      = lane >> 4;       // 0 for lanes 0-15, 1 for lanes 16-31

    const long blk_tok0 = (long)blockIdx.x * TOK_PER_BLK;

    // B fragments (4x16 f32): lane_lo selects N-col, hi selects K pair {0,1} vs {2,3}.
    const float* bptr0 = Wr + (long)( 0 + lane_lo) * HDIM + 2 * hi;
    const float* bptr1 = Wr + (long)(16 + lane_lo) * HDIM + 2 * hi;
    const float* bptr2 = Wr + (long)(32 + lane_lo) * HDIM + 2 * hi;
    const float* bptr3 = Wr + (long)(48 + lane_lo) * HDIM + 2 * hi;

    v8f c0 = {}, c1 = {}, c2 = {}, c3 = {};

#if HAVE_ASYNC_LDS
    // ---- Async double-buffered A staging: 128 rows x 64 floats per chunk ----
    // 2048 x b128 per chunk, 256 threads x 8 issues; rows padded to stride 65
    // (each b128 stays within one row since rows are 16 x 16B).
    auto issue_chunk = [&](int buf, int kc) {
        #pragma unroll
        for (int it = 0; it < 8; ++it) {
            const int q   = it * 256 + tid;
            const int row = q >> 4;      // 16 x b128 per row
            const int c16 = q & 15;
            const float* g = hidden + (blk_tok0 + row) * HDIM + kc + c16 * 4;
            float* l = &smem[buf][row * LDS_STRIDE + c16 * 4];
            __builtin_amdgcn_global_load_async_to_lds_b128(
                (GLOBAL_AS v4i*)g,       // addrspacecast: global src
                (LDS_AS v4i*)l,          // addrspacecast: LDS dst
                /*imm offset*/0, /*cpol*/0);
        }
    };

    issue_chunk(0, 0);
    // A fragment base inside a staged chunk for this lane.
    const int arow_off = (wave * TOK_PER_WAVE + lane_lo) * LDS_STRIDE + 2 * hi;

    for (int c = 0; c < HDIM / KCHUNK; ++c) {
        const int cur = c & 1;
        if (c + 1 < HDIM / KCHUNK) {
            issue_chunk(cur ^ 1, (c + 1) * KCHUNK);
            WAIT_ASYNC(8);               // chunk c landed (in-order completion)
        } else {
            WAIT_ASYNC(0);
        }
        __syncthreads();

        const float* arow  = &smem[cur][arow_off];
        const int    kbase = c * KCHUNK;
        #pragma unroll 4
        for (int kk = 0; kk < KCHUNK; kk += 4) {
            v2f a  = *(const v2f*)(arow + kk);               // ds_load_b64, conflict-free
            v2f b0 = *(const v2f*)(bptr0 + kbase + kk);
            v2f b1 = *(const v2f*)(bptr1 + kbase + kk);
            v2f b2 = *(const v2f*)(bptr2 + kbase + kk);
            v2f b3 = *(const v2f*)(bptr3 + kbase + kk);
            c0 = __builtin_amdgcn_wmma_f32_16x16x4_f32(false, a, false, b0, (short)0, c0, false, false);
            c1 = __builtin_amdgcn_wmma_f32_16x16x4_f32(false, a, false, b1, (short)0, c1, false, false);
            c2 = __builtin_amdgcn_wmma_f32_16x16x4_f32(false, a, false, b2, (short)0, c2, false, false);
            c3 = __builtin_amdgcn_wmma_f32_16x16x4_f32(false, a, false, b3, (short)0, c3, false, false);
        }
        __syncthreads();                 // all waves done reading buf[cur] before reuse
    }
#else
    // ---- Fallback: direct per-lane global b64 A loads (proven path) ----
    const float* aptr = hidden + (blk_tok0 + wave * TOK_PER_WAVE + lane_lo) * HDIM + 2 * hi;
    #pragma unroll 4
    for (int k = 0; k < HDIM; k += 4) {
        if ((k & 255) == 0) {            // uniform branch: EXEC untouched
            __builtin_prefetch(aptr + k + 512, 0, 3);   // near/WGP-scope prefetch
        }
        v2f a  = *(const v2f*)(aptr  + k);
        v2f b0 = *(const v2f*)(bptr0 + k);
        v2f b1 = *(const v2f*)(bptr1 + k);
        v2f b2 = *(const v2f*)(bptr2 + k);
        v2f b3 = *(const v2f*)(bptr3 + k);
        c0 = __builtin_amdgcn_wmma_f32_16x16x4_f32(false, a, false, b0, (short)0, c0, false, false);
        c1 = __builtin_amdgcn_wmma_f32_16x16x4_f32(false, a, false, b1, (short)0, c1, false, false);
        c2 = __builtin_amdgcn_wmma_f32_16x16x4_f32(false, a, false, b2, (short)0, c2, false, false);
        c3 = __builtin_amdgcn_wmma_f32_16x16x4_f32(false, a, false, b3, (short)0, c3, false, false);
    }
    __syncthreads();                     // smem[0] free for epilogue staging
#endif

    // C layout: lanes 0-15 -> N=lane_lo, M=j; lanes 16-31 -> N=lane_lo, M=j+8.
    float* lstage = &smem[0][0];
    #pragma unroll
    for (int j = 0; j < 8; ++j) {
        const int trow = wave * TOK_PER_WAVE + j + 8 * hi;
        float* r = lstage + trow * LDS_STRIDE + lane_lo;
        r[ 0] = c0[j];
        r[16] = c1[j];
        r[32] = c2[j];
        r[48] = c3[j];
    }
    __syncthreads();

    // Coalesced logits writeback (float4).
    for (int i = tid; i < TOK_PER_BLK * NEXP / 4; i += 256) {
        const int le = i * 4;
        const int t  = le >> 6;          // / 64
        const int e0 = le & 63;
        const float* r = lstage + t * LDS_STRIDE + e0;
        v4f v = { r[0], r[1], r[2], r[3] };
        *(v4f*)(out_logits + (blk_tok0 + t) * NEXP + e0) = v;
    }

    // Per-token top-2 + renormalized softmax weights (denominator cancels:
    // w1 = 1/(1+exp(l2-l1)), w2 = 1-w1).
    if (tid < TOK_PER_BLK) {
        const float* row = lstage + tid * LDS_STRIDE;
        float m1 = row[0]; int i1 = 0;
        float m2 = -3.402823466e38f; int i2 = 0;
        #pragma unroll 8
        for (int e = 1; e < NEXP; ++e) {
            const float v = row[e];
            if (v > m1)      { m2 = m1; i2 = i1; m1 = v; i1 = e; }
            else if (v > m2) { m2 = v; i2 = e; }
        }
        const float w1 = 1.0f / (1.0f + __expf(m2 - m1));   // m2 - m1 <= 0
        const long t = blk_tok0 + tid;
        out_wflat[t * 2 + 0] = w1;
        out_wflat[t * 2 + 1] = 1.0f - w1;
        idx_ws[t * 2 + 0] = i1;
        idx_ws[t * 2 + 1] = i2;
    }
}

// One wave per expert: zero capacity tables, then ordered slot-major scan with
// ballot/prefix-popcount reproducing the reference's greedy assignment exactly.
__global__ __launch_bounds__(32) void assign_experts_kernel(
    const int* __restrict__ idx_ws,      // [T, 2]
    const float* __restrict__ wflat,     // [T, 2]
    int* __restrict__ out_idx,           // [NEXP, cap]
    float* __restrict__ out_w,           // [NEXP, cap]
    int T, int cap)
{
    const int e    = blockIdx.x;
    const int lane = threadIdx.x;

    for (int i = lane; i < cap; i += 32) {
        out_idx[e * cap + i] = 0;
        out_w  [e * cap + i] = 0.0f;
    }

    int count = 0;
    const unsigned long long lane_mask = (1ull << lane) - 1ull;
    for (int base = 0; base < 2 * T && count < cap; base += 32) {
        const int o     = base + lane;           // slot-major order index
        const int slot  = (o >= T) ? 1 : 0;
        const int token = o - slot * T;
        const int ex    = idx_ws[token * 2 + slot];
        const bool m    = (ex == e);
        const unsigned long long mask = __ballot(m);
        if (m) {
            const int pos = count + __popcll(mask & lane_mask);
            if (pos < cap) {
                out_idx[e * cap + pos] = token;
                out_w  [e * cap + pos] = wflat[token * 2 + slot];
            }
        }
        count += __popcll(mask);
    }
}

extern "C" void kernel_launch(void* const* d_in, const int* in_sizes, int n_in,
                              void* d_out, int out_size, void* d_ws, size_t ws_size,
                              hipStream_t stream) {
    const float* hidden = (const float*)d_in[0];   // [B,S,H] f32
    const float* Wr     = (const float*)d_in[1];   // [E,H]   f32

    const int T   = in_sizes[0] / HDIM;            // 16384
    const int cap = (T / NEXP) * 5 / 4;            // int(T/E * 1.25) = 320

    int*   out_idx    = (int*)d_out;               // E*cap i32 (bit-stored)
    float* out_w      = (float*)d_out + (long)NEXP * cap;
    float* out_wflat  = out_w + (long)NEXP * cap;
    float* out_logits = out_wflat + (long)T * 2;
    int*   idx_ws     = (int*)d_ws;                // [T,2] scratch

    router_gemm_topk_kernel<<<dim3(T / TOK_PER_BLK), 256, 0, stream>>>(
        hidden, Wr, out_logits, out_wflat, idx_ws);
    assign_experts_kernel<<<dim3(NEXP), 32, 0, stream>>>(
        idx_ws, out_wflat, out_idx, out_w, T, cap);
}